// Net_52329881534570
// MI455X (gfx1250) — compile-verified
//
#include <hip/hip_runtime.h>
#include <stdint.h>

#define DIM_FEAT 5
#define CHUNK    2048
#define BLOCK    256

typedef __attribute__((ext_vector_type(4))) unsigned int v4u;
typedef __attribute__((ext_vector_type(8))) int          v8i;
typedef __attribute__((ext_vector_type(4))) int          v4i;

// ---------------------------------------------------------------------------
// TDM: DMA `nelem` 4-byte elements from global `gaddr` into LDS at byte
// offset `lds_off`, as a 1-row tile (tile_dim0 = nelem, tile_dim1 = 1).
// Descriptor layout per CDNA5 ISA.
// This toolchain (clang-23 / therock-10.0) exposes the 6-arg builtin.
// ---------------------------------------------------------------------------
__device__ __forceinline__ void tdm_load_1d_b32(uint32_t lds_off,
                                                uint64_t gaddr,
                                                uint32_t nelem) {
  v4u g0;
  g0.x = 1u;                                      // count=1, user mode, no gather
  g0.y = lds_off;                                 // lds_addr (bytes)
  g0.z = (uint32_t)(gaddr & 0xFFFFFFFFull);       // global_addr[31:0]
  g0.w = (uint32_t)((gaddr >> 32) & 0x01FFFFFFull)// global_addr[56:32]
       | (2u << 30);                              // type = 2 ("image")

  v8i g1;
  g1[0] = (int)(2u << 16);                        // wg_mask=0, data_size=2 (4B)
  g1[1] = (int)((nelem & 0xFFFFu) << 16);         // abar_addr=0 | tensor_dim0[15:0]
  g1[2] = (int)(((nelem >> 16) & 0xFFFFu)         // tensor_dim0[31:16]
       |  (1u << 16));                            // tensor_dim1[15:0] = 1
  g1[3] = (int)((nelem & 0xFFFFu) << 16);         // tensor_dim1[31:16]=0 | tile_dim0=n
  g1[4] = 1;                                      // tile_dim1=1, tile_dim2=0
  g1[5] = (int)nelem;                             // tensor_dim0_stride[31:0]
  g1[6] = (int)((nelem >> 16) & 0xFFFFu);         // stride[47:32] | dim1_stride[15:0]=0
  g1[7] = 0;

  v4i z4 = {0, 0, 0, 0};                          // groups 2/3: dims 2..4 unused
  v8i z8 = {0, 0, 0, 0, 0, 0, 0, 0};              // extra group (zero-filled)
  __builtin_amdgcn_tensor_load_to_lds(g0, g1, z4, z4, z8, 0);
}

__device__ __forceinline__ void wait_tensorcnt0() {
#if defined(__has_builtin) && __has_builtin(__builtin_amdgcn_s_wait_tensorcnt)
  __builtin_amdgcn_s_wait_tensorcnt(0);
#else
  asm volatile("s_wait_tensorcnt 0" ::: "memory");
#endif
}

// ---------------------------------------------------------------------------
// Zero the output. Must run every launch: the harness poisons d_out once and
// graph-replays accumulate otherwise.
// ---------------------------------------------------------------------------
__global__ void zero_out_kernel(float* __restrict__ out, int n) {
  int i = blockIdx.x * blockDim.x + threadIdx.x;
  if (i < n) out[i] = 0.0f;
}

// ---------------------------------------------------------------------------
// One 2048-edge chunk per block. Wave 0 TDM-stages the src/dst index chunks
// into LDS, waits on TENSORcnt, then all 8 waves gather feature rows (L2-hot,
// 2 MB table) and scatter with agent-scope relaxed float atomics (L2-hot,
// 2 MB output).
// ---------------------------------------------------------------------------
__global__ void gnn_scatter_kernel(const int* __restrict__ src,
                                   const int* __restrict__ dst,
                                   const float* __restrict__ feat,
                                   float* __restrict__ out,
                                   int n_edges) {
  __shared__ int s_src[CHUNK];
  __shared__ int s_dst[CHUNK];

  const int base = blockIdx.x * CHUNK;
  const int n    = min(CHUNK, n_edges - base);
  if (n <= 0) return;

  if (threadIdx.x < 32) {  // wave 0 only: TDM issues once per wave
    tdm_load_1d_b32((uint32_t)(uintptr_t)s_src,
                    (uint64_t)(uintptr_t)(src + base), (uint32_t)n);
    tdm_load_1d_b32((uint32_t)(uintptr_t)s_dst,
                    (uint64_t)(uintptr_t)(dst + base), (uint32_t)n);
    wait_tensorcnt0();
  }
  __syncthreads();

  for (int e = threadIdx.x; e < n; e += BLOCK) {
    const int s = s_src[e];
    const int d = s_dst[e];
    const float* __restrict__ frow = feat + (size_t)s * DIM_FEAT;
    float* __restrict__       orow = out  + (size_t)d * DIM_FEAT;
#pragma unroll
    for (int f = 0; f < DIM_FEAT; ++f) {
      __hip_atomic_fetch_add(&orow[f], frow[f],
                             __ATOMIC_RELAXED, __HIP_MEMORY_SCOPE_AGENT);
    }
  }
}

extern "C" void kernel_launch(void* const* d_in, const int* in_sizes, int n_in,
                              void* d_out, int out_size, void* d_ws, size_t ws_size,
                              hipStream_t stream) {
  const float* feat = (const float*)d_in[0];   // [N_NODES, 5] f32
  const int*   src  = (const int*)d_in[1];     // [E] int
  const int*   dst  = (const int*)d_in[2];     // [E] int
  float*       out  = (float*)d_out;           // [N_NODES, 5] f32
  const int n_edges = in_sizes[1];

  zero_out_kernel<<<(out_size + 255) / 256, 256, 0, stream>>>(out, out_size);

  const int nblocks = (n_edges + CHUNK - 1) / CHUNK;
  gnn_scatter_kernel<<<nblocks, BLOCK, 0, stream>>>(src, dst, feat, out, n_edges);
}